// ContLoss_9045201125869
// MI455X (gfx1250) — compile-verified
//
#include <hip/hip_runtime.h>
#include <math.h>

typedef __attribute__((ext_vector_type(2))) float v2f;
typedef __attribute__((ext_vector_type(8))) float v8f;

#define T_LEN    512
#define B_SZ     64
#define E_SZ     1024
#define N_NEG    128
#define INV_TEMP 10.0f
#define CL_EPS   1e-8f

// ---------------------------------------------------------------------------
// Kernel 1: positive-pair cosine for 16 timesteps per wave via f32 WMMA.
// D_AB = A * B^T accumulated over E in K=4 chunks; diag(D_AB)[m] = dot(orig_t, adv_t),
// diag(D_AA) = ||orig||^2, diag(D_BB) = ||adv||^2.
// 32 threads = 1 wave32, EXEC all-ones as WMMA requires.
// ---------------------------------------------------------------------------
__global__ __launch_bounds__(32)
void ContLoss_pos_wmma(const int* __restrict__ index,
                       const float* __restrict__ z1,
                       const float* __restrict__ z2,
                       float* __restrict__ cos_out) {
    const int lane = threadIdx.x;            // 0..31
    const int t0   = blockIdx.x * 16;
    const int row  = lane & 15;              // M (for A) == N (for B)
    const int kq   = (lane >> 4) << 1;       // K base: 0 for lanes 0-15, 2 for 16-31
    const int t    = t0 + row;
    const int bi   = index[t];

    const float* ap = z1 + ((size_t)t * B_SZ + bi) * E_SZ;  // orig row
    const float* bp = z2 + ((size_t)t * B_SZ + bi) * E_SZ;  // adv  row

    v8f cab = {};  // orig . adv
    v8f caa = {};  // orig . orig
    v8f cbb = {};  // adv  . adv

    #pragma unroll 4
    for (int e0 = 0; e0 < E_SZ; e0 += 4) {
        const v2f a = *(const v2f*)(ap + e0 + kq);   // A[16x4] f32 layout
        const v2f b = *(const v2f*)(bp + e0 + kq);   // B[4x16] f32 layout (symmetric fetch)
        cab = __builtin_amdgcn_wmma_f32_16x16x4_f32(false, a, false, b,
                                                    (short)0, cab, false, false);
        caa = __builtin_amdgcn_wmma_f32_16x16x4_f32(false, a, false, a,
                                                    (short)0, caa, false, false);
        cbb = __builtin_amdgcn_wmma_f32_16x16x4_f32(false, b, false, b,
                                                    (short)0, cbb, false, false);
    }

    // Diagonal extraction. C/D layout (32-bit 16x16): lane L<16 holds N=L with
    // M = vgpr (0..7); lane L>=16 holds N=L-16 with M = vgpr+8.
    // -> diagonal on lanes 0-7 (vgpr = lane) and lanes 24-31 (vgpr = lane-24).
    const bool has_diag = (lane < 8) || (lane >= 24);
    const int  c = lane & 7;
    float dab = cab[0], daa = caa[0], dbb = cbb[0];
    #pragma unroll
    for (int i = 1; i < 8; ++i) {
        if (c == i) { dab = cab[i]; daa = caa[i]; dbb = cbb[i]; }
    }

    if (has_diag) {
        const int tout = t0 + ((lane < 8) ? lane : (lane - 16));
        const float na = fmaxf(sqrtf(daa), CL_EPS);
        const float nb = fmaxf(sqrtf(dbb), CL_EPS);
        cos_out[tout] = dab / (na * nb);
    }
}

// ---------------------------------------------------------------------------
// Kernel 2: negative term. One block per t. 256 threads, thread owns 4 e's.
// Streams 128 random 4KB rows with coalesced B128 loads + software prefetch,
// accumulating sum and sum-of-squares over N, then computes
// den[t] = sum_e exp(10 * cos_neg[t,e]) and the per-t loss.
// ---------------------------------------------------------------------------
__global__ __launch_bounds__(256)
void ContLoss_neg_kernel(const int* __restrict__ index,
                         const float* __restrict__ z1,
                         const int* __restrict__ neg_sent,
                         const int* __restrict__ neg_word,
                         const float* __restrict__ cos_pos,
                         float* __restrict__ loss_out) {
    __shared__ unsigned s_off[N_NEG];   // element offsets of the 128 negative rows
    __shared__ float    s_red[256];

    const int t   = blockIdx.x;
    const int tid = threadIdx.x;

    if (tid < N_NEG) {
        const unsigned rs = (unsigned)neg_sent[t * N_NEG + tid];
        const unsigned rw = (unsigned)neg_word[t * N_NEG + tid];
        s_off[tid] = (rs * B_SZ + rw) * E_SZ;
    }
    __syncthreads();

    const int e0 = tid * 4;             // 16B-aligned within each 4KB row
    float4 s = make_float4(0.f, 0.f, 0.f, 0.f);
    float4 q = make_float4(0.f, 0.f, 0.f, 0.f);

    #pragma unroll 4
    for (int n = 0; n < N_NEG; ++n) {
        if (n + 8 < N_NEG)
            __builtin_prefetch(z1 + s_off[n + 8] + e0, 0, 0);  // global_prefetch_b8
        const float4 v = *(const float4*)(z1 + s_off[n] + e0); // global_load_b128
        s.x += v.x; s.y += v.y; s.z += v.z; s.w += v.w;
        q.x = fmaf(v.x, v.x, q.x);
        q.y = fmaf(v.y, v.y, q.y);
        q.z = fmaf(v.z, v.z, q.z);
        q.w = fmaf(v.w, v.w, q.w);
    }

    // anchor row for this t
    const int bi = index[t];
    const float4 o = *(const float4*)(z1 + ((size_t)t * B_SZ + bi) * E_SZ + e0);

    const float sqrtN = 11.313708498984761f;  // sqrt(128)
    float part = 0.f;
    {
        float nn, no, cs;
        nn = fmaxf(sqrtf(q.x), CL_EPS); no = fmaxf(sqrtN * fabsf(o.x), CL_EPS);
        cs = (o.x * s.x) / (nn * no);   part += __expf(cs * INV_TEMP);
        nn = fmaxf(sqrtf(q.y), CL_EPS); no = fmaxf(sqrtN * fabsf(o.y), CL_EPS);
        cs = (o.y * s.y) / (nn * no);   part += __expf(cs * INV_TEMP);
        nn = fmaxf(sqrtf(q.z), CL_EPS); no = fmaxf(sqrtN * fabsf(o.z), CL_EPS);
        cs = (o.z * s.z) / (nn * no);   part += __expf(cs * INV_TEMP);
        nn = fmaxf(sqrtf(q.w), CL_EPS); no = fmaxf(sqrtN * fabsf(o.w), CL_EPS);
        cs = (o.w * s.w) / (nn * no);   part += __expf(cs * INV_TEMP);
    }

    // deterministic fixed-order tree reduction of den over the block
    s_red[tid] = part;
    __syncthreads();
    #pragma unroll
    for (int off = 128; off > 0; off >>= 1) {
        if (tid < off) s_red[tid] += s_red[tid + off];
        __syncthreads();
    }

    if (tid == 0) {
        // -log(pos/den) = log(den) - cos_pos/TEMP
        loss_out[t] = __logf(s_red[0]) - cos_pos[t] * INV_TEMP;
    }
}

// ---------------------------------------------------------------------------
// Kernel 3: deterministic reduction of the 512 per-t losses to the scalar.
// ---------------------------------------------------------------------------
__global__ __launch_bounds__(256)
void ContLoss_final_reduce(const float* __restrict__ loss_in,
                           float* __restrict__ out) {
    __shared__ float s[256];
    const int tid = threadIdx.x;
    s[tid] = loss_in[tid] + loss_in[tid + 256];
    __syncthreads();
    #pragma unroll
    for (int off = 128; off > 0; off >>= 1) {
        if (tid < off) s[tid] += s[tid + off];
        __syncthreads();
    }
    if (tid == 0) out[0] = s[0];
}

// ---------------------------------------------------------------------------
extern "C" void kernel_launch(void* const* d_in, const int* in_sizes, int n_in,
                              void* d_out, int out_size, void* d_ws, size_t ws_size,
                              hipStream_t stream) {
    const int*   index    = (const int*)d_in[0];
    const float* z1       = (const float*)d_in[1];
    const float* z2       = (const float*)d_in[2];
    const int*   neg_sent = (const int*)d_in[3];
    const int*   neg_word = (const int*)d_in[4];

    float* ws      = (float*)d_ws;
    float* cos_pos = ws;           // [T_LEN]
    float* loss_pt = ws + T_LEN;   // [T_LEN]
    float* out     = (float*)d_out;

    ContLoss_pos_wmma<<<T_LEN / 16, 32, 0, stream>>>(index, z1, z2, cos_pos);
    ContLoss_neg_kernel<<<T_LEN, 256, 0, stream>>>(index, z1, neg_sent, neg_word,
                                                   cos_pos, loss_pt);
    ContLoss_final_reduce<<<1, 256, 0, stream>>>(loss_pt, out);
}